// AttentionBlock64_12850542149908
// MI455X (gfx1250) — compile-verified
//
#include <hip/hip_runtime.h>

typedef float v2f __attribute__((ext_vector_type(2)));
typedef float v4f __attribute__((ext_vector_type(4)));
typedef float v8f __attribute__((ext_vector_type(8)));
typedef int   v4i __attribute__((ext_vector_type(4)));

#define LDS_STRIDE 68   // 64 + 4 floats pad -> conflict-free fragment fetches
#define AS1 __attribute__((address_space(1)))
#define AS3 __attribute__((address_space(3)))

// ---- CDNA5 async global->LDS path (ASYNCcnt), with compile-safe fallback ----
#if __has_builtin(__builtin_amdgcn_global_load_async_to_lds_b128)
#define ASYNC_LDS 1
#else
#define ASYNC_LDS 0
#endif

#if __has_builtin(__builtin_amdgcn_s_wait_asynccnt)
#define WAIT_ASYNC(n) __builtin_amdgcn_s_wait_asynccnt(n)
#else
#define WAIT_ASYNC(n) asm volatile("s_wait_asynccnt " #n ::: "memory")
#endif

__device__ __forceinline__ void tile_copy_issue(const float* __restrict__ gsrc,
                                                float* __restrict__ dst,
                                                int t, size_t base)
{
    // one 64x64 f32 tile = 1024 float4 slots; 256 threads -> 4 per thread
    #pragma unroll
    for (int i = 0; i < 4; ++i) {
        const int idx = t + i * 256;
        const int row = idx >> 4;
        const int c4  = idx & 15;
#if ASYNC_LDS
        __builtin_amdgcn_global_load_async_to_lds_b128(
            (AS1 v4i*)(gsrc + base + row * 64 + c4 * 4),
            (AS3 v4i*)(dst + row * LDS_STRIDE + c4 * 4),
            /*offset=*/0, /*cpol=*/0);
#else
        v4f val = *(const v4f*)(gsrc + base + row * 64 + c4 * 4);
        *(v4f*)(dst + row * LDS_STRIDE + c4 * 4) = val;
#endif
    }
}

__global__ __launch_bounds__(256)
void attn64_wmma_f32_kernel(const float* __restrict__ q,
                            const float* __restrict__ k,
                            const float* __restrict__ v,
                            const float* __restrict__ proj,
                            const float* __restrict__ residual,
                            float* __restrict__ out)
{
    __shared__ float smQ[64 * LDS_STRIDE];
    __shared__ float smK[64 * LDS_STRIDE];   // K, then CTX
    __shared__ float smV[64 * LDS_STRIDE];
    __shared__ float smP[64 * LDS_STRIDE];
    __shared__ float smE[64 * LDS_STRIDE];   // E = exp(S)
    __shared__ float inv_rowsum[64];

    const int b    = blockIdx.x;
    const int t    = threadIdx.x;
    const int lane = t & 31;
    const int wave = t >> 5;                 // 8 waves
    const int tn   = wave & 3;               // tile col (0..3)
    const int wr   = wave >> 2;              // tile rows {wr, wr+2}

    const size_t base = (size_t)b * 64 * 64;

    // fragment indexing for V_WMMA_F32_16X16X4_F32
    //   A(16x4): lane -> m = lane%16 ; VGPR j -> k = 2*(lane/16)+j
    //   B(4x16): lane -> n = lane%16 ; VGPR j -> k = 2*(lane/16)+j
    //   C/D:     lane -> n = lane%16 ; VGPR i -> m = i + 8*(lane/16)
    const int mA = lane & 15;
    const int kA = (lane >> 4) * 2;
    const int mC = (lane >> 4) * 8;
    const int nC = lane & 15;

    const int rowA0 = (wr * 16      + mA) * LDS_STRIDE;  // A-frag row, tile tm=wr
    const int rowA1 = (wr * 16 + 32 + mA) * LDS_STRIDE;  // A-frag row, tile tm=wr+2
    const int colB  = tn * 16 + nC;

    // ---- issue ALL tile loads up front (async path streams behind compute) ----
    tile_copy_issue(q,    smQ, t, base);     // ops 1..4   (per thread)
    tile_copy_issue(k,    smK, t, base);     // ops 5..8
    tile_copy_issue(v,    smV, t, base);     // ops 9..12
    tile_copy_issue(proj, smP, t, base);     // ops 13..16
    // warm L2 for the epilogue residual read (256 x 256B covers the tile)
    __builtin_prefetch(residual + base + (size_t)t * 64, 0, 3);

#if ASYNC_LDS
    WAIT_ASYNC(8);                           // Q and K arrived (in-order completion)
#endif
    __syncthreads();

    // ---------------- Phase 1: S = Q @ K^T, E = exp(S) ----------------
    v8f acc0 = {0.f,0.f,0.f,0.f,0.f,0.f,0.f,0.f};
    v8f acc1 = {0.f,0.f,0.f,0.f,0.f,0.f,0.f,0.f};
    #pragma unroll
    for (int kk = 0; kk < 16; ++kk) {
        const int kb = kk * 4 + kA;
        // B frag: B[d][j] = K[j][d] -> K row (tn*16+n), cols kb..kb+1 (shared by both tiles)
        v2f bf = *(const v2f*)(smK + (tn * 16 + mA) * LDS_STRIDE + kb);
        v2f a0 = *(const v2f*)(smQ + rowA0 + kb);
        v2f a1 = *(const v2f*)(smQ + rowA1 + kb);
        acc0 = __builtin_amdgcn_wmma_f32_16x16x4_f32(false, a0, false, bf, (short)0, acc0, false, false);
        acc1 = __builtin_amdgcn_wmma_f32_16x16x4_f32(false, a1, false, bf, (short)0, acc1, false, false);
    }

    // exp in registers (faithful: no max subtraction), spill E tiles to LDS
    #pragma unroll
    for (int i = 0; i < 8; ++i) {
        acc0[i] = __expf(acc0[i]);
        acc1[i] = __expf(acc1[i]);
        smE[(wr * 16      + mC + i) * LDS_STRIDE + colB] = acc0[i];
        smE[(wr * 16 + 32 + mC + i) * LDS_STRIDE + colB] = acc1[i];
    }

    __syncthreads();   // E complete; Q,K dead

    // inverse row sums of E (threads 0..63); V already streamed into smV
    if (t < 64) {
        float s = 0.f;
        #pragma unroll
        for (int j = 0; j < 64; ++j) s += smE[t * LDS_STRIDE + j];
        inv_rowsum[t] = 1.0f / s;
    }
#if ASYNC_LDS
    WAIT_ASYNC(0);                           // V and PROJ arrived
#endif
    __syncthreads();

    // ---------------- Phase 2: CTX = diag(1/rowsum) * (E @ V) ----------------
    acc0 = (v8f){0.f,0.f,0.f,0.f,0.f,0.f,0.f,0.f};
    acc1 = (v8f){0.f,0.f,0.f,0.f,0.f,0.f,0.f,0.f};
    #pragma unroll
    for (int kk = 0; kk < 16; ++kk) {
        const int kb = kk * 4 + kA;
        v2f bf;                               // B = V row-major: (k,n) = V[k][n]
        bf.x = smV[(kb + 0) * LDS_STRIDE + colB];
        bf.y = smV[(kb + 1) * LDS_STRIDE + colB];
        v2f a0 = *(const v2f*)(smE + rowA0 + kb);
        v2f a1 = *(const v2f*)(smE + rowA1 + kb);
        acc0 = __builtin_amdgcn_wmma_f32_16x16x4_f32(false, a0, false, bf, (short)0, acc0, false, false);
        acc1 = __builtin_amdgcn_wmma_f32_16x16x4_f32(false, a1, false, bf, (short)0, acc1, false, false);
    }

    // fold softmax normalization into ctx rows; stash CTX into smK (dead)
    #pragma unroll
    for (int i = 0; i < 8; ++i) {
        acc0[i] *= inv_rowsum[wr * 16      + mC + i];
        acc1[i] *= inv_rowsum[wr * 16 + 32 + mC + i];
        smK[(wr * 16      + mC + i) * LDS_STRIDE + colB] = acc0[i];
        smK[(wr * 16 + 32 + mC + i) * LDS_STRIDE + colB] = acc1[i];
    }

    __syncthreads();   // CTX complete; E,V dead

    // ---------------- Phase 3: OUT = CTX @ PROJ + residual ----------------
    acc0 = (v8f){0.f,0.f,0.f,0.f,0.f,0.f,0.f,0.f};
    acc1 = (v8f){0.f,0.f,0.f,0.f,0.f,0.f,0.f,0.f};
    #pragma unroll
    for (int kk = 0; kk < 16; ++kk) {
        const int kb = kk * 4 + kA;
        v2f bf;                               // B = PROJ row-major
        bf.x = smP[(kb + 0) * LDS_STRIDE + colB];
        bf.y = smP[(kb + 1) * LDS_STRIDE + colB];
        v2f a0 = *(const v2f*)(smK + rowA0 + kb);
        v2f a1 = *(const v2f*)(smK + rowA1 + kb);
        acc0 = __builtin_amdgcn_wmma_f32_16x16x4_f32(false, a0, false, bf, (short)0, acc0, false, false);
        acc1 = __builtin_amdgcn_wmma_f32_16x16x4_f32(false, a1, false, bf, (short)0, acc1, false, false);
    }

    // fused residual add + store, per-lane global access at C/D layout
    #pragma unroll
    for (int i = 0; i < 8; ++i) {
        const size_t g0 = base + (size_t)(wr * 16      + mC + i) * 64 + colB;
        const size_t g1 = base + (size_t)(wr * 16 + 32 + mC + i) * 64 + colB;
        out[g0] = acc0[i] + residual[g0];
        out[g1] = acc1[i] + residual[g1];
    }
}

extern "C" void kernel_launch(void* const* d_in, const int* in_sizes, int n_in,
                              void* d_out, int out_size, void* d_ws, size_t ws_size,
                              hipStream_t stream) {
    const float* q        = (const float*)d_in[0];
    const float* k        = (const float*)d_in[1];
    const float* v        = (const float*)d_in[2];
    const float* proj     = (const float*)d_in[3];
    const float* residual = (const float*)d_in[4];
    float* out            = (float*)d_out;

    const int num_tiles = in_sizes[0] / (64 * 64);   // B = 8192
    attn64_wmma_f32_kernel<<<num_tiles, 256, 0, stream>>>(q, k, v, proj, residual, out);
}